// SpikingFEASTNetwork_71940702208581
// MI455X (gfx1250) — compile-verified
//
#include <hip/hip_runtime.h>

// ---------------------------------------------------------------------------
// Collapsed FEAST network:
//   c1 = argmin_n ||w1[n]-x||^2  == argmin_n ( sum w1[n]^2 - 2*w1[n].x )
//   c2 = argmin_n ||w2[n]-e_c1||^2 == argmin_n ( sum w2[n]^2 - 2*w2[n][c1] )
//   out[o] = fcw[o][c2] + fcb[o]
// Memory-bound: ~256 MB of weight reads -> ~11 us floor at 23.3 TB/s.
// Layer-1 dot products computed with V_WMMA_F32_16X16X4_F32.
// ---------------------------------------------------------------------------

typedef __attribute__((ext_vector_type(2))) float v2f;
typedef __attribute__((ext_vector_type(8))) float v8f;

#define D1     4096   // x dimension
#define H1     8192   // layer-1 neurons (rows of w1)
#define H2     4096   // layer-2 neurons (rows of w2)
#define OUTN   512    // fc outputs
#define KSPLIT 4      // waves per 16-row group (K-dimension split)
#define W2_WAVES 8    // waves per block in layer-2 kernel

// Order-preserving float -> uint mapping (total order, -inf..+inf ascending).
__device__ __forceinline__ unsigned int order_f32(float f) {
  unsigned int u = __float_as_uint(f);
  return (u & 0x80000000u) ? ~u : (u | 0x80000000u);
}

__global__ void init_ws_kernel(unsigned long long* __restrict__ ws) {
  if (threadIdx.x < 2) ws[threadIdx.x] = ~0ull;
}

// ---------------------------------------------------------------------------
// Layer 1: one block per 16-row group of w1; KSPLIT waves split the K axis.
// Per wave:  A fragment = 16x4 f32 tile of w1  (one B64 global load per lane)
//            B fragment = x chunk broadcast across all 16 columns (ds_load_b64)
//            C += A*B  accumulates w.x replicated across columns
//            sumsq accumulated with VALU FMAs from the same registers.
// ---------------------------------------------------------------------------
__global__ __launch_bounds__(32 * KSPLIT)
void feast1_wmma_kernel(const float* __restrict__ w1,
                        const float* __restrict__ x,
                        unsigned long long* __restrict__ best1) {
  __shared__ float xs[D1];
  __shared__ float part[KSPLIT][16];

  for (int i = threadIdx.x; i < D1; i += blockDim.x) xs[i] = x[i];
  __syncthreads();

  const int lane    = threadIdx.x & 31;
  const int wave    = threadIdx.x >> 5;       // which K-slice
  const int rowBase = blockIdx.x * 16;
  const int l       = lane & 15;              // row within group
  const int hi      = lane >> 4;              // 0: K%4 in {0,1}; 1: K%4 in {2,3}
  const int k0      = wave * (D1 / KSPLIT);
  const int k1      = k0 + (D1 / KSPLIT);

  const float* wrow = w1 + (size_t)(rowBase + l) * D1 + 2 * hi;
  const float* xsl  = xs + 2 * hi;

  v8f  c  = {};       // dot(w_row, x) accumulators, replicated over N columns
  float sq = 0.0f;    // partial sum of squares for row l (this lane's K subset)

  for (int k = k0; k < k1; k += 4) {
    v2f a = *(const v2f*)(wrow + k);   // A: w[row][k+2hi], w[row][k+2hi+1]
    v2f b = *(const v2f*)(xsl  + k);   // B: x[k+2hi], x[k+2hi+1] (bcast over N)
    sq = __builtin_fmaf(a.x, a.x, sq);
    sq = __builtin_fmaf(a.y, a.y, sq);
    // D = A x B + C : 16x16x4 f32
    c = __builtin_amdgcn_wmma_f32_16x16x4_f32(false, a, false, b,
                                              (short)0, c, false, false);
  }

  // Combine sumsq halves (K mod 4 = {0,1} lanes with {2,3} lanes).
  sq += __shfl_xor(sq, 16);

  // C/D layout: VGPR v holds row v (lanes 0-15) and row v+8 (lanes 16-31),
  // replicated across columns since B was column-broadcast.
  // Owner lanes: lane r (r<8) -> row r ; lane 16+r -> row 8+r.
  union { v8f v; float f[8]; } cu; cu.v = c;
  if (lane < 8 || (lane >= 16 && lane < 24)) {
    const int r = (lane < 8) ? lane : (lane - 16 + 8);
    part[wave][r] = sq - 2.0f * cu.f[lane & 7];  // partial (||w||^2 - 2 w.x)
  }
  __syncthreads();

  // Deterministic fixed-order combine over K-slices, then block argmin.
  if (threadIdx.x < 16) {
    const int r = threadIdx.x;
    float val = 0.0f;
#pragma unroll
    for (int wv = 0; wv < KSPLIT; ++wv) val += part[wv][r];
    unsigned long long key =
        ((unsigned long long)order_f32(val) << 32) |
        (unsigned int)(rowBase + r);
#pragma unroll
    for (int off = 8; off > 0; off >>= 1) {
      unsigned long long o = __shfl_xor(key, off);
      key = (o < key) ? o : key;
    }
    if (threadIdx.x == 0) atomicMin(best1, key);
  }
}

// ---------------------------------------------------------------------------
// Layer 2: value[n] = sum(w2[n]^2) - 2*w2[n][c1]  (const +1 dropped).
// Wave-per-row, float4 loads -> perfectly coalesced B128, 4096 waves.
// ---------------------------------------------------------------------------
__global__ __launch_bounds__(32 * W2_WAVES)
void feast2_kernel(const float* __restrict__ w2,
                   const unsigned long long* __restrict__ best1,
                   unsigned long long* __restrict__ best2) {
  __shared__ unsigned long long keys[W2_WAVES];

  const int lane = threadIdx.x & 31;
  const int wave = threadIdx.x >> 5;
  const int row  = blockIdx.x * W2_WAVES + wave;
  const unsigned int c1 = (unsigned int)(*best1 & 0xffffffffull);

  const float4* p = (const float4*)(w2 + (size_t)row * H1);
  float sq = 0.0f;
#pragma unroll 4
  for (int i = lane; i < H1 / 4; i += 32) {
    float4 v = p[i];
    sq = __builtin_fmaf(v.x, v.x, sq);
    sq = __builtin_fmaf(v.y, v.y, sq);
    sq = __builtin_fmaf(v.z, v.z, sq);
    sq = __builtin_fmaf(v.w, v.w, sq);
  }
#pragma unroll
  for (int off = 16; off > 0; off >>= 1) sq += __shfl_xor(sq, off);

  if (lane == 0) {
    const float val = sq - 2.0f * w2[(size_t)row * H1 + c1];
    keys[wave] = ((unsigned long long)order_f32(val) << 32) | (unsigned int)row;
  }
  __syncthreads();

  if (threadIdx.x == 0) {
    unsigned long long k = keys[0];
#pragma unroll
    for (int wv = 1; wv < W2_WAVES; ++wv) k = (keys[wv] < k) ? keys[wv] : k;
    atomicMin(best2, k);
  }
}

// ---------------------------------------------------------------------------
// FC: out[o] = fcw[o][c2] + fcb[o]  (column gather of fcw).
// ---------------------------------------------------------------------------
__global__ void fc_gather_kernel(const float* __restrict__ fcw,
                                 const float* __restrict__ fcb,
                                 const unsigned long long* __restrict__ best2,
                                 float* __restrict__ out) {
  const unsigned int c2 = (unsigned int)(*best2 & 0xffffffffull);
  const int o = blockIdx.x * blockDim.x + threadIdx.x;
  if (o < OUTN) out[o] = fcw[(size_t)o * H2 + c2] + fcb[o];
}

extern "C" void kernel_launch(void* const* d_in, const int* in_sizes, int n_in,
                              void* d_out, int out_size, void* d_ws, size_t ws_size,
                              hipStream_t stream) {
  const float* x   = (const float*)d_in[0];
  // d_in[1] = reward : unused by the reference network
  const float* w1  = (const float*)d_in[2];
  // d_in[3] = t1     : does not affect argmin -> unused
  const float* w2  = (const float*)d_in[4];
  // d_in[5] = t2     : unused
  const float* fcw = (const float*)d_in[6];
  const float* fcb = (const float*)d_in[7];
  float* out = (float*)d_out;

  unsigned long long* ws = (unsigned long long*)d_ws;  // ws[0]=best1, ws[1]=best2

  hipLaunchKernelGGL(init_ws_kernel, dim3(1), dim3(32), 0, stream, ws);
  hipLaunchKernelGGL(feast1_wmma_kernel, dim3(H1 / 16), dim3(32 * KSPLIT), 0,
                     stream, w1, x, &ws[0]);
  hipLaunchKernelGGL(feast2_kernel, dim3(H2 / W2_WAVES), dim3(32 * W2_WAVES), 0,
                     stream, w2, &ws[0], &ws[1]);
  hipLaunchKernelGGL(fc_gather_kernel, dim3((OUTN + 255) / 256), dim3(256), 0,
                     stream, fcw, fcb, &ws[1], out);
}